// Manifold_30477087933290
// MI455X (gfx1250) — compile-verified
//
#include <hip/hip_runtime.h>
#include <hip/hip_bf16.h>
#include <stdint.h>

#define V 8192
#define F 16384

// ---------------------------------------------------------------------------
// Kernel 1: zero-fill 256 MB of output through the CDNA5 async store engine.
// Each lane repeatedly issues GLOBAL_STORE_ASYNC_FROM_LDS_B128 (ASYNCcnt path,
// fire-and-forget: no STOREcnt stall), sourcing 16 B of zeros from LDS.
// TH_STORE_NT: output (256 MB) exceeds L2 (192 MB) -> stream, don't pollute.
// ---------------------------------------------------------------------------
__global__ void fill_zero_async(float* __restrict__ out) {
    __shared__ alignas(16) float zbuf[64];          // 256 B of zeros, 16 chunks
    if (threadIdx.x < 64) zbuf[threadIdx.x] = 0.0f;
    __syncthreads();                                 // ds stores visible to TDM/async path

    // LDS byte address for this lane (spread over banks by lane%16)
    const unsigned lds_off =
        (unsigned)(uintptr_t)(&zbuf[0]) + (threadIdx.x & 15u) * 16u;

    const size_t lanes = (size_t)gridDim.x * blockDim.x;           // 524288
    size_t idx = (size_t)blockIdx.x * blockDim.x + threadIdx.x;
    unsigned long long addr =
        (unsigned long long)(uintptr_t)out + idx * 16ull;
    const unsigned long long stride = lanes * 16ull;               // 8 MB

    #pragma unroll 4
    for (int i = 0; i < 32; ++i) {                   // 32 * 524288 * 16B = 256 MB
        asm volatile(
            "global_store_async_from_lds_b128 %0, %1, off th:TH_STORE_NT scope:SCOPE_DEV"
            :: "v"(addr), "v"(lds_off) : "memory");
        addr += stride;
    }
    asm volatile("s_wait_asynccnt 0x0" ::: "memory");
}

// ---------------------------------------------------------------------------
// Kernel 2: one thread per edge-slot (3F = 49152). Compute the cotan weight
// and scatter-add 4 entries with hardware global_atomic_add_f32.
// ---------------------------------------------------------------------------
__global__ void cotan_scatter(const float* __restrict__ fs,
                              const int*   __restrict__ faces,
                              float*       __restrict__ L) {
    const int e = blockIdx.x * blockDim.x + threadIdx.x;
    if (e >= 3 * F) return;
    const int f = e / 3;
    const int j = e - 3 * f;
    const int j1 = (j + 1 == 3) ? 0 : j + 1;
    const int j2 = (j + 2 >= 3) ? (j + 2 - 3) : j + 2;

    const int a = faces[3 * f + j ];   // tail
    const int b = faces[3 * f + j1];   // opposite vertex (angle here)
    const int c = faces[3 * f + j2];   // tip

    const float ax = fs[3*a+0], ay = fs[3*a+1], az = fs[3*a+2];
    const float bx = fs[3*b+0], by = fs[3*b+1], bz = fs[3*b+2];
    const float cx = fs[3*c+0], cy = fs[3*c+1], cz = fs[3*c+2];

    // e_kj = b - a ; e_ki = b - c   (angle at b, opposite edge a<->c)
    const float ux = bx-ax, uy = by-ay, uz = bz-az;
    const float vx = bx-cx, vy = by-cy, vz = bz-cz;

    const float d  = ux*vx + uy*vy + uz*vz;
    const float nn = sqrtf((ux*ux + uy*uy + uz*uz) * (vx*vx + vy*vy + vz*vz));
    float cosa = d / nn;
    cosa = fminf(1.0f, fmaxf(-1.0f, cosa));
    // 0.5 / tan(acos(cosa)) == 0.5 * cosa / sqrt(1 - cosa^2)
    const float sina = sqrtf(fmaxf(1.0f - cosa * cosa, 0.0f));
    const float w = 0.5f * cosa / sina;

    unsafeAtomicAdd(L + (size_t)a * V + c,  w);
    unsafeAtomicAdd(L + (size_t)c * V + a,  w);
    unsafeAtomicAdd(L + (size_t)a * V + a, -w);
    unsafeAtomicAdd(L + (size_t)c * V + c, -w);
}

extern "C" void kernel_launch(void* const* d_in, const int* in_sizes, int n_in,
                              void* d_out, int out_size, void* d_ws, size_t ws_size,
                              hipStream_t stream) {
    const float* fs    = (const float*)d_in[0];   // (V,3) float32
    const int*   faces = (const int*)  d_in[1];   // (F,3) int32
    float*       L     = (float*)d_out;           // (V,V) float32

    // 2048 blocks * 256 threads * 32 chunks * 16 B == 256 MB exactly
    fill_zero_async<<<2048, 256, 0, stream>>>(L);
    cotan_scatter<<<(3 * F + 255) / 256, 256, 0, stream>>>(fs, faces, L);
}